// PagedAttention_67422396613021
// MI455X (gfx1250) — compile-verified
//
#include <hip/hip_runtime.h>
#include <hip/hip_bf16.h>

// ---------------- problem constants (match reference) ----------------
#define NUM_SEQS   32
#define NUM_HEADS  32
#define NUM_KV     8
#define QPG        4          // query heads per kv head (GQA)
#define HEAD_D     128
#define BS         16         // paged block size == WMMA N tile
#define MAXB       128        // blocks per sequence
#define MAXCTX     2048
#define ATTN_SCALE 0.08838834764831845f   // 1/sqrt(128)

typedef __attribute__((ext_vector_type(2))) float v2f;
typedef __attribute__((ext_vector_type(8))) float v8f;

// One workgroup = one (seq, kv_head). 8 waves (wave32) per workgroup.
__global__ __launch_bounds__(256, 2)
void PagedAttention_wmma_kernel(const float* __restrict__ query,
                                const float* __restrict__ key_cache,
                                const float* __restrict__ value_cache,
                                const int*   __restrict__ block_tables,
                                const int*   __restrict__ context_lens,
                                float*       __restrict__ out)
{
    __shared__ float sQ[QPG][HEAD_D];      // 2 KB   query rows for this kv head
    __shared__ float sP[QPG][MAXCTX];      // 32 KB  scores -> exp(scores - max)
    __shared__ float sRed[256];            // 1 KB   reduction scratch
    __shared__ float rowMax[QPG];
    __shared__ float rowSum[QPG];

    const int g    = blockIdx.x;           // kv head
    const int s    = blockIdx.y;           // sequence
    const int tid  = threadIdx.x;
    const int wave = tid >> 5;
    const int lane = tid & 31;
    const int m    = lane & 15;            // M (A/C row group) / N column index
    const int kh   = lane >> 4;            // K-half select for 16x16x4 frags

    const int L       = context_lens[s];
    const int nTiles  = (L + BS - 1) / BS; // one cache block per tile
    const int* btab   = block_tables + s * MAXB;

    // ---- stage Q rows for this kv-head group into LDS ----
    for (int i = tid; i < QPG * HEAD_D; i += 256) {
        const int q = i >> 7;              // /128
        const int d = i & (HEAD_D - 1);
        sQ[q][d] = query[(size_t)s * (NUM_HEADS * HEAD_D) + (g * QPG + q) * HEAD_D + d];
    }
    __syncthreads();

    // A-fragments for Q (constant across key tiles): 32 K-chunks of 4 over D=128.
    // A[m][k] layout: lanes 0-15 hold K={0,1}, lanes 16-31 hold K={2,3}.
    v2f aq[32];
    #pragma unroll
    for (int c = 0; c < 32; ++c) {
        const int d0 = 4 * c + 2 * kh;
        if (m < QPG) { aq[c].x = sQ[m][d0]; aq[c].y = sQ[m][d0 + 1]; }
        else         { aq[c].x = 0.0f;      aq[c].y = 0.0f; }
    }

    // ================= Phase 1: scores = (Q K^T) * scale  =================
    // key_cache element [blk][g][d/8][bs][x]; per (blk,g) slab = 16*16*8 floats
    for (int t = wave; t < nTiles; t += 8) {
        const int blk = btab[t];
        const float* kb = key_cache + ((size_t)blk * NUM_KV + g) * (16 * BS * 8);

        if (t + 8 < nTiles) {
            const int nblk = btab[t + 8];
            __builtin_prefetch(key_cache + ((size_t)nblk * NUM_KV + g) * (16 * BS * 8), 0, 1);
        }

        v8f acc = {};
        #pragma unroll
        for (int c = 0; c < 32; ++c) {
            const int d0 = 4 * c + 2 * kh;         // even => pair stays inside X=8 group
            // B[k][n]: n = key position within block = m; dims d0, d0+1 contiguous
            const float* kp = kb + (d0 >> 3) * (BS * 8) + m * 8 + (d0 & 7);
            v2f b; b.x = kp[0]; b.y = kp[1];
            acc = __builtin_amdgcn_wmma_f32_16x16x4_f32(
                      false, aq[c], false, b, (short)0, acc, false, false);
        }
        // C/D: lanes 0-15 VGPR r holds row M=r (rows 0..3 are the real queries)
        if (kh == 0) {
            const int col = t * BS + m;
            sP[0][col] = acc[0] * ATTN_SCALE;
            sP[1][col] = acc[1] * ATTN_SCALE;
            sP[2][col] = acc[2] * ATTN_SCALE;
            sP[3][col] = acc[3] * ATTN_SCALE;
        }
    }
    __syncthreads();

    // ================= Phase 2: softmax (normalization deferred) ==========
    {
        const int q  = tid >> 6;           // 4 rows x 64 threads
        const int ln = tid & 63;

        float mx = -__builtin_inff();
        for (int j = ln; j < MAXCTX; j += 64)
            if (j < L) mx = fmaxf(mx, sP[q][j]);
        sRed[tid] = mx;
        __syncthreads();
        for (int off = 32; off > 0; off >>= 1) {
            if (ln < off) sRed[tid] = fmaxf(sRed[tid], sRed[tid + off]);
            __syncthreads();
        }
        if (ln == 0) rowMax[q] = sRed[tid];
        __syncthreads();

        const float rmax = rowMax[q];
        float sum = 0.0f;
        for (int j = ln; j < MAXCTX; j += 64) {
            float e = 0.0f;
            if (j < L) e = __expf(sP[q][j] - rmax);
            sP[q][j] = e;                  // zero beyond L -> safe for phase 3
            sum += e;
        }
        sRed[tid] = sum;
        __syncthreads();
        for (int off = 32; off > 0; off >>= 1) {
            if (ln < off) sRed[tid] += sRed[tid + off];
            __syncthreads();
        }
        if (ln == 0) rowSum[q] = sRed[tid];
    }
    __syncthreads();

    // ================= Phase 3: O = P V  (wave owns a 16-wide D slice) =====
    // value_cache element [blk][g][d][bs]; per (blk,g) slab = 128*16 floats
    const int dt = wave;                   // d-tile: dims dt*16 .. dt*16+15
    v8f acc = {};
    for (int t = 0; t < nTiles; ++t) {
        const int blk = btab[t];
        const float* vb = value_cache + ((size_t)blk * NUM_KV + g) * (HEAD_D * BS);

        if (t + 1 < nTiles) {
            const int nblk = btab[t + 1];
            __builtin_prefetch(value_cache + ((size_t)nblk * NUM_KV + g) * (HEAD_D * BS), 0, 1);
        }

        #pragma unroll
        for (int cc = 0; cc < 4; ++cc) {   // 4 K-chunks of 4 positions = 16
            v2f a;
            if (m < QPG) {
                const int col = t * BS + cc * 4 + 2 * kh;
                a.x = sP[m][col]; a.y = sP[m][col + 1];
            } else { a.x = 0.0f; a.y = 0.0f; }
            // B[k][n]: n = d offset = m; positions bs0, bs0+1 contiguous (bs fastest)
            const float* vp = vb + (dt * 16 + m) * BS + cc * 4 + 2 * kh;
            v2f b; b.x = vp[0]; b.y = vp[1];
            acc = __builtin_amdgcn_wmma_f32_16x16x4_f32(
                      false, a, false, b, (short)0, acc, false, false);
        }
    }

    // epilogue: normalize by softmax denominator and store
    if (kh == 0) {
        float* op = out + (size_t)s * (NUM_HEADS * HEAD_D)
                        + (g * QPG) * HEAD_D + dt * 16 + m;
        op[0 * HEAD_D] = acc[0] / rowSum[0];
        op[1 * HEAD_D] = acc[1] / rowSum[1];
        op[2 * HEAD_D] = acc[2] / rowSum[2];
        op[3 * HEAD_D] = acc[3] / rowSum[3];
    }
}

extern "C" void kernel_launch(void* const* d_in, const int* in_sizes, int n_in,
                              void* d_out, int out_size, void* d_ws, size_t ws_size,
                              hipStream_t stream) {
    const float* query       = (const float*)d_in[0];
    const float* key_cache   = (const float*)d_in[1];
    const float* value_cache = (const float*)d_in[2];
    const int*   block_tables = (const int*)d_in[3];
    const int*   context_lens = (const int*)d_in[4];
    float* out = (float*)d_out;

    dim3 grid(NUM_KV, NUM_SEQS);   // (kv head, sequence)
    PagedAttention_wmma_kernel<<<grid, 256, 0, stream>>>(
        query, key_cache, value_cache, block_tables, context_lens, out);
}